// LBMSolver_42563125903664
// MI455X (gfx1250) — compile-verified
//
#include <hip/hip_runtime.h>
#include <stdint.h>

// ---------------------------------------------------------------------------
// D2Q9 LBM step (collide -> stream -> moments), fused single kernel.
// Memory-bound (AI ~1.5 FLOP/B << machine balance at 23.3 TB/s): optimize the
// data path, not the math. gfx1250 features used:
//   * global_load_async_to_lds_b32 (ASYNCcnt) DMAs the f halo tile to LDS
//     with no VGPR round-trip
//   * staged s_wait_asynccnt waits software-pipeline collision against DMA
//     (async loads complete in order within a wave, ISA 7.1)
//   * 34x34x9 f32 LDS tile (41.6 KB of 320 KB/WGP); stride-9-dword layout is
//     bank-conflict-free for 32 lanes (gcd(9,64)=1)
//   * non-temporal stores for the ~400 MB streamed output (keep the 100 MB
//     rho/u arrays resident in the 192 MB L2)
// WMMA is deliberately unused: the only contractions (c.u, moments) are K=2 /
// K=9 and lane-local; WMMA would need cross-lane transposes costing more than
// the 18 VALU adds they replace.
// ---------------------------------------------------------------------------

#define NYD 2048
#define NXD 4096
#define TILE 32
#define HALO 34                    // TILE + 2 (one-cell periodic halo)
#define NHALO (HALO * HALO)        // 1156 halo cells
#define NTHREADS 256
#define NITER ((NHALO + NTHREADS - 1) / NTHREADS)   // 5

__device__ __forceinline__ void collide9(const float fi[9], float r, float ux,
                                         float uy, float fo[9]) {
  const float omega = 1.0f / 0.6f;            // 1/TAU
  const float usq15 = 1.5f * (ux * ux + uy * uy);
  float cu[9];
  cu[0] = 0.0f;
  cu[1] = ux;        cu[2] = uy;
  cu[3] = -ux;       cu[4] = -uy;
  cu[5] = ux + uy;   cu[6] = uy - ux;
  cu[7] = -ux - uy;  cu[8] = ux - uy;
  const float w[9] = {4.0f / 9.0f,  1.0f / 9.0f,  1.0f / 9.0f,
                      1.0f / 9.0f,  1.0f / 9.0f,  1.0f / 36.0f,
                      1.0f / 36.0f, 1.0f / 36.0f, 1.0f / 36.0f};
#pragma unroll
  for (int q = 0; q < 9; ++q) {
    float c = cu[q];
    float feq = w[q] * r * (1.0f + 3.0f * c + 4.5f * c * c - usq15);
    fo[q] = fi[q] + (feq - fi[q]) * omega;
  }
}

// One async b32 per dword; the ISA adds INST_OFFSET to BOTH the LDS address
// (VDST vgpr) and the global address (SADDR + VADDR vgpr), so the 9 dwords of
// one cell share a single (laddr, voff) pair.
#define ALD(OFF)                                                               \
  asm volatile("global_load_async_to_lds_b32 %0, %1, %2 offset:" #OFF          \
               :: "v"(laddr), "v"(voff), "s"(fbase) : "memory")

// Wait until at most N of this wave's async loads remain outstanding.
#define AWAIT(N) asm volatile("s_wait_asynccnt " #N ::: "memory")

__global__ __launch_bounds__(NTHREADS) void lbm_d2q9_step(
    const float* __restrict__ f, const float* __restrict__ rho,
    const float* __restrict__ u, float* __restrict__ fn,
    float* __restrict__ rhon, float* __restrict__ un) {
  __shared__ float fst[NHALO * 9];  // 41616 B: raw f, then f_star in place

  const int tid = threadIdx.x;
  const int x0 = blockIdx.x * TILE;
  const int y0 = blockIdx.y * TILE;

  // Low 32 bits of a generic pointer to LDS == LDS byte offset (aperture enc).
  const uint32_t lds_base = (uint32_t)(uintptr_t)fst;
  const unsigned long long fbase = (unsigned long long)f;

  // ---- Phase 0: async-stage the 34x34 halo tile of f into LDS -------------
  int gidx[NITER];
  float r_[NITER], ux_[NITER], uy_[NITER];
#pragma unroll
  for (int k = 0; k < NITER; ++k) {
    const int c = tid + k * NTHREADS;
    const bool valid = (c < NHALO);          // only batch k=4 can be invalid
    const int cc = valid ? c : (NHALO - 1);
    const int hy = cc / HALO;
    const int hx = cc - hy * HALO;
    const int gy = (y0 + hy - 1) & (NYD - 1);   // periodic wrap (pow2 dims)
    const int gx = (x0 + hx - 1) & (NXD - 1);
    const int gi = gy * NXD + gx;
    gidx[k] = gi;
    if (valid) {
      const uint32_t voff = (uint32_t)gi * 36u;              // f byte offset
      const uint32_t laddr = lds_base + (uint32_t)cc * 36u;  // LDS byte offset
      ALD(0); ALD(4); ALD(8); ALD(12); ALD(16); ALD(20); ALD(24); ALD(28); ALD(32);
    }
  }
  // Overlap: pull L2-resident rho/u into registers while the DMA is in flight.
#pragma unroll
  for (int k = 0; k < NITER; ++k) {
    const int gi = gidx[k];
    r_[k] = rho[gi];
    ux_[k] = u[2 * gi + 0];
    uy_[k] = u[2 * gi + 1];
  }

  // ---- Phase 1: BGK collision in place, pipelined against the DMA ---------
  // Each thread collides exactly the cells its own wave DMA'd, and async
  // loads complete in order, so no block barrier is needed here. The wait
  // schedule (27,18,9,0,0) is valid whether a wave issued 45 loads or
  // branch-skipped the last batch and issued 36.
  auto collide_batch = [&](int k) {
    const int c = tid + k * NTHREADS;
    if (c < NHALO) {
      float fi[9], fo[9];
#pragma unroll
      for (int q = 0; q < 9; ++q) fi[q] = fst[c * 9 + q];
      collide9(fi, r_[k], ux_[k], uy_[k], fo);
#pragma unroll
      for (int q = 0; q < 9; ++q) fst[c * 9 + q] = fo[q];
    }
  };
  AWAIT(27); collide_batch(0);
  AWAIT(18); collide_batch(1);
  AWAIT(9);  collide_batch(2);
  AWAIT(0);  collide_batch(3);
  collide_batch(4);

  __syncthreads();   // single barrier: f_star tile complete for all threads

  // ---- Phase 2: streaming gather from LDS + moments + NT writeback --------
  const int cx9[9] = {0, 1, 0, -1, 0, 1, -1, -1, 1};
  const int cy9[9] = {0, 0, 1, 0, -1, 1, 1, -1, -1};
#pragma unroll
  for (int m = 0; m < (TILE * TILE) / NTHREADS; ++m) {
    const int i = tid + m * NTHREADS;  // 0..1023
    const int ty = i >> 5;
    const int tx = i & 31;
    float f9[9];
#pragma unroll
    for (int q = 0; q < 9; ++q) {
      const int sy = ty + 1 - cy9[q];  // pull scheme: src = dst - c_q
      const int sx = tx + 1 - cx9[q];
      f9[q] = fst[(sy * HALO + sx) * 9 + q];  // stride-9 dwords: conflict-free
    }
    const float rn = ((f9[0] + f9[1]) + (f9[2] + f9[3])) +
                     ((f9[4] + f9[5]) + (f9[6] + f9[7])) + f9[8];
    const float mx = (f9[1] - f9[3]) + (f9[5] - f9[6]) + (f9[8] - f9[7]);
    const float my = (f9[2] - f9[4]) + (f9[5] + f9[6]) - (f9[7] + f9[8]);
    const float inv = 1.0f / rn;

    const size_t gi = (size_t)(y0 + ty) * NXD + (size_t)(x0 + tx);
    float* __restrict__ fo = fn + gi * 9;
#pragma unroll
    for (int q = 0; q < 9; ++q) __builtin_nontemporal_store(f9[q], fo + q);
    __builtin_nontemporal_store(rn, rhon + gi);
    __builtin_nontemporal_store(mx * inv, un + 2 * gi + 0);
    __builtin_nontemporal_store(my * inv, un + 2 * gi + 1);
  }
}

extern "C" void kernel_launch(void* const* d_in, const int* in_sizes, int n_in,
                              void* d_out, int out_size, void* d_ws,
                              size_t ws_size, hipStream_t stream) {
  (void)in_sizes; (void)n_in; (void)out_size; (void)d_ws; (void)ws_size;
  const float* f = (const float*)d_in[0];    // (NY, NX, 9)
  const float* rho = (const float*)d_in[1];  // (NY, NX)
  const float* u = (const float*)d_in[2];    // (NY, NX, 2)

  float* out = (float*)d_out;                // f_new | rho_new | u_new
  float* fn = out;
  float* rhon = fn + (size_t)NYD * NXD * 9;
  float* un = rhon + (size_t)NYD * NXD;

  dim3 grid(NXD / TILE, NYD / TILE);  // 128 x 64
  dim3 block(NTHREADS);
  hipLaunchKernelGGL(lbm_d2q9_step, grid, block, 0, stream, f, rho, u, fn,
                     rhon, un);
}